// constraints_enforcement_41936060678216
// MI455X (gfx1250) — compile-verified
//
#include <hip/hip_runtime.h>
#include <hip/hip_bf16.h>
#include <math.h>
#include <stdint.h>

#ifndef __has_builtin
#define __has_builtin(x) 0
#endif

#if defined(__gfx1250__) &&                                              \
    __has_builtin(__builtin_amdgcn_global_load_async_to_lds_b128) &&     \
    __has_builtin(__builtin_amdgcn_global_store_async_from_lds_b128)
#define USE_ASYNC_LDS 1
#else
#define USE_ASYNC_LDS 0
#endif

#if USE_ASYNC_LDS
#define AS1 __attribute__((address_space(1)))
#define AS3 __attribute__((address_space(3)))
typedef int v4i __attribute__((vector_size(4 * sizeof(int))));
typedef AS1 v4i* gptr;  // global-memory pointer to 128b chunk
typedef AS3 v4i* lptr;  // LDS pointer to 128b chunk

#if __has_builtin(__builtin_amdgcn_s_wait_asynccnt)
#define ASYNC_WAIT0() __builtin_amdgcn_s_wait_asynccnt(0)
#else
#define ASYNC_WAIT0() asm volatile("s_wait_asynccnt 0" ::: "memory")
#endif
#endif  // USE_ASYNC_LDS

#define TPB 256

// Rodrigues rotation aligning v1 -> v2, matching the JAX reference
// (including degenerate sin and antipodal branches).
__device__ __forceinline__ void rot_between(float x0, float x1, float x2,
                                            float y0, float y1, float y2,
                                            float* R) {
  const float n1 = sqrtf(x0 * x0 + x1 * x1 + x2 * x2);
  const float n2 = sqrtf(y0 * y0 + y1 * y1 + y2 * y2);
  const float a0 = x0 / n1, a1 = x1 / n1, a2 = x2 / n1;
  const float b0 = y0 / n2, b1 = y1 / n2, b2 = y2 / n2;

  const float v0 = a1 * b2 - a2 * b1;
  const float v1 = a2 * b0 - a0 * b2;
  const float v2 = a0 * b1 - a1 * b0;
  const float c  = a0 * b0 + a1 * b1 + a2 * b2;
  const float s2 = v0 * v0 + v1 * v1 + v2 * v2;
  const float s  = sqrtf(s2);

  // K*K = v v^T - s2 * I  ->  R = I + K + coef*(v v^T - s2*I)
  const float s2s  = (s2 > 0.0f) ? s2 : 1.0f;
  const float coef = (1.0f - c) / s2s;

  R[0] = 1.0f + coef * (v0 * v0 - s2);
  R[1] = -v2  + coef * (v0 * v1);
  R[2] =  v1  + coef * (v0 * v2);
  R[3] =  v2  + coef * (v0 * v1);
  R[4] = 1.0f + coef * (v1 * v1 - s2);
  R[5] = -v0  + coef * (v1 * v2);
  R[6] = -v1  + coef * (v0 * v2);
  R[7] =  v0  + coef * (v1 * v2);
  R[8] = 1.0f + coef * (v2 * v2 - s2);

  if (s < 1e-30f) {
    if (c > 0.0f) {
      // identity branch
      R[0] = 1.0f; R[1] = 0.0f; R[2] = 0.0f;
      R[3] = 0.0f; R[4] = 1.0f; R[5] = 0.0f;
      R[6] = 0.0f; R[7] = 0.0f; R[8] = 1.0f;
    } else if (c < 0.0f) {
      // antipodal (180 deg) branch
      const float T0  = 1.0e-8f + 1.0e-5f;  // ATOL + RTOL*|1|
      const float T12 = 1.0e-8f;            // ATOL + RTOL*|0|
      const bool close_e1 = (fabsf(a0 - 1.0f) <= T0) &&
                            (fabsf(a1) <= T12) && (fabsf(a2) <= T12);
      const float ax0 = close_e1 ? 0.0f : 1.0f;  // axis = close_e1 ? e2 : e1
      const float ax1 = close_e1 ? 1.0f : 0.0f;
      // perp = cross(a, axis) with axis.z == 0
      float p0 = -a2 * ax1;
      float p1 =  a2 * ax0;
      float p2 =  a0 * ax1 - a1 * ax0;
      const float pn = sqrtf(p0 * p0 + p1 * p1 + p2 * p2);
      const float pd = (pn > 0.0f) ? pn : 1.0f;
      p0 /= pd; p1 /= pd; p2 /= pd;
      const float qq = p0 * p0 + p1 * p1 + p2 * p2;
      // R180 = I + 2*(p p^T - (p.p) I)
      R[0] = 1.0f + 2.0f * (p0 * p0 - qq);
      R[1] = 2.0f * p0 * p1;
      R[2] = 2.0f * p0 * p2;
      R[3] = 2.0f * p0 * p1;
      R[4] = 1.0f + 2.0f * (p1 * p1 - qq);
      R[5] = 2.0f * p1 * p2;
      R[6] = 2.0f * p0 * p2;
      R[7] = 2.0f * p1 * p2;
      R[8] = 1.0f + 2.0f * (p2 * p2 - qq);
    }
    // s_zero && c == 0: keep Rodrigues result (masks false in reference)
  }
}

__global__ __launch_bounds__(TPB) void rodrigues_align_kernel(
    const float* __restrict__ v1g, const float* __restrict__ v2g,
    float* __restrict__ outg, int npairs) {
  __shared__ __align__(16) float aIn[TPB * 3];   // 3 KB
  __shared__ __align__(16) float bIn[TPB * 3];   // 3 KB
  __shared__ __align__(16) float rOut[TPB * 9];  // 9 KB

  const int tid = threadIdx.x;
  const long long base = (long long)blockIdx.x * TPB;

  if (base + TPB <= (long long)npairs) {
    // ---- full tile: LDS-staged, fully coalesced 128b path ----
    const float* g1 = v1g + base * 3;
    const float* g2 = v2g + base * 3;

#if USE_ASYNC_LDS
    if (tid < (TPB * 3) / 4) {  // 192 lanes x 16B = 3KB per buffer
      __builtin_amdgcn_global_load_async_to_lds_b128(
          (gptr)(g1 + tid * 4), (lptr)(aIn + tid * 4), 0, 0);
      __builtin_amdgcn_global_load_async_to_lds_b128(
          (gptr)(g2 + tid * 4), (lptr)(bIn + tid * 4), 0, 0);
    }
    ASYNC_WAIT0();
    __syncthreads();
#else
#pragma unroll
    for (int k = 0; k < 3; ++k) {
      aIn[k * TPB + tid] = g1[k * TPB + tid];
      bIn[k * TPB + tid] = g2[k * TPB + tid];
    }
    __syncthreads();
#endif

    float R[9];
    rot_between(aIn[tid * 3 + 0], aIn[tid * 3 + 1], aIn[tid * 3 + 2],
                bIn[tid * 3 + 0], bIn[tid * 3 + 1], bIn[tid * 3 + 2], R);
#pragma unroll
    for (int k = 0; k < 9; ++k) rOut[tid * 9 + k] = R[k];
    __syncthreads();

    float* go = outg + base * 9;
#if USE_ASYNC_LDS
    // 9KB tile = 576 x b128 chunks: every lane does 2, lanes < 64 do a 3rd
    __builtin_amdgcn_global_store_async_from_lds_b128(
        (gptr)(go + tid * 4), (lptr)(rOut + tid * 4), 0, 0);
    __builtin_amdgcn_global_store_async_from_lds_b128(
        (gptr)(go + (TPB + tid) * 4), (lptr)(rOut + (TPB + tid) * 4), 0, 0);
    if (tid < (TPB * 9) / 4 - 2 * TPB) {
      __builtin_amdgcn_global_store_async_from_lds_b128(
          (gptr)(go + (2 * TPB + tid) * 4), (lptr)(rOut + (2 * TPB + tid) * 4),
          0, 0);
    }
    ASYNC_WAIT0();
#else
#pragma unroll
    for (int k = 0; k < 9; ++k) go[k * TPB + tid] = rOut[k * TPB + tid];
#endif
  } else {
    // ---- tail (block-uniform branch; no barriers here) ----
    for (long long i = base + tid; i < (long long)npairs; i += TPB) {
      float R[9];
      rot_between(v1g[i * 3 + 0], v1g[i * 3 + 1], v1g[i * 3 + 2],
                  v2g[i * 3 + 0], v2g[i * 3 + 1], v2g[i * 3 + 2], R);
#pragma unroll
      for (int k = 0; k < 9; ++k) outg[i * 9 + k] = R[k];
    }
  }
}

extern "C" void kernel_launch(void* const* d_in, const int* in_sizes, int n_in,
                              void* d_out, int out_size, void* d_ws,
                              size_t ws_size, hipStream_t stream) {
  const float* v1 = (const float*)d_in[0];
  const float* v2 = (const float*)d_in[1];
  float* out = (float*)d_out;
  const int npairs = in_sizes[0] / 3;  // (B, NB, 3) flattened
  const int blocks = (npairs + TPB - 1) / TPB;
  rodrigues_align_kernel<<<blocks, TPB, 0, stream>>>(v1, v2, out, npairs);
}